// SimpleHGNConv_81552839016950
// MI455X (gfx1250) — compile-verified
//
#include <hip/hip_runtime.h>
#include <math.h>

#define HEADS 4
#define OUT_DIM 32
#define IN_DIM 128
#define EDGE_DIM 32
#define NEG_SLOPE 0.2f

typedef float v2f __attribute__((ext_vector_type(2)));
typedef float v8f __attribute__((ext_vector_type(8)));

#define WMMA_F32(a, b, c)                                                     \
    __builtin_amdgcn_wmma_f32_16x16x4_f32(false, (a), false, (b), (short)0,   \
                                          (c), false, false)

// ---------------------------------------------------------------------------
// float atomic max via int/uint trick (buffer must be initialized to -inf)
// ---------------------------------------------------------------------------
__device__ __forceinline__ void atomicMaxF(float* addr, float v) {
    if (v >= 0.0f)
        atomicMax((int*)addr, __float_as_int(v));
    else
        atomicMin((unsigned int*)addr, __float_as_uint(v));
}

// ---------------------------------------------------------------------------
// 0) init: zero the output accumulator and den, set segment-max to -inf
// ---------------------------------------------------------------------------
__global__ void init_kernel(float* __restrict__ out, float* __restrict__ mbuf,
                            float* __restrict__ den, int n_nodes) {
    int idx = blockIdx.x * blockDim.x + threadIdx.x;
    if (idx < n_nodes * (OUT_DIM * HEADS)) out[idx] = 0.0f;
    if (idx < n_nodes * HEADS) {
        mbuf[idx] = -INFINITY;
        den[idx]  = 0.0f;
    }
}

// ---------------------------------------------------------------------------
// 1) fp32 WMMA GEMM: emb = h @ W ; res = h @ res_w   (both N x 128, K = 128)
//    2x2 register tiling: each wave32 computes a 32x32 output patch as four
//    16x16 tiles via V_WMMA_F32_16X16X4_F32 (4 wmma per K-step, A and B
//    fragments each reused twice). block = 128 (4 waves);
//    grid = (ceil(N/32), 2); col-tile pair = blockIdx.y*4 + waveId in [0,8):
//    pairs [0,4) -> W/emb, [4,8) -> res_w/res (wave-uniform select).
// ---------------------------------------------------------------------------
__global__ void gemm_proj_kernel(const float* __restrict__ h,
                                 const float* __restrict__ W,
                                 const float* __restrict__ resw,
                                 float* __restrict__ emb,
                                 float* __restrict__ res,
                                 int n_nodes) {
    const int lane   = threadIdx.x & 31;
    const int waveId = threadIdx.x >> 5;
    const int pair   = blockIdx.y * 4 + waveId;   // 0..7 (wave-uniform)
    const int tile_m = blockIdx.x;                // 32-row group

    const float* Bm;
    float*       Out;
    int          ncol0;
    if (pair < 4) { Bm = W;    Out = emb; ncol0 = pair * 32; }
    else          { Bm = resw; Out = res; ncol0 = (pair - 4) * 32; }

    const int m    = lane & 15;     // A row / B-C-D col within tile
    const int half = lane >> 4;     // K-pair select for A/B fragments

    const int r0  = tile_m * 32 + m;
    const int r1  = r0 + 16;
    const int r0c = (r0 < n_nodes) ? r0 : (n_nodes - 1);
    const int r1c = (r1 < n_nodes) ? r1 : (n_nodes - 1);

    const float* hr0 = h + (size_t)r0c * IN_DIM;
    const float* hr1 = h + (size_t)r1c * IN_DIM;

    v8f acc00 = {}, acc01 = {}, acc10 = {}, acc11 = {};
    for (int k = 0; k < IN_DIM; k += 4) {
        const int ka = k + 2 * half;
        v2f a0, a1, b0, b1;
        // A 16x4 fp32 fragments: lanes 0-15 K={k,k+1}, lanes 16-31 K={k+2,k+3}
        a0.x = hr0[ka];  a0.y = hr0[ka + 1];
        a1.x = hr1[ka];  a1.y = hr1[ka + 1];
        // B 4x16 fp32 fragments (N striped across lanes)
        const float* Bk0 = Bm + (size_t)ka * 128 + ncol0 + m;
        const float* Bk1 = Bm + (size_t)(ka + 1) * 128 + ncol0 + m;
        b0.x = Bk0[0];   b0.y = Bk1[0];
        b1.x = Bk0[16];  b1.y = Bk1[16];
        acc00 = WMMA_F32(a0, b0, acc00);
        acc01 = WMMA_F32(a0, b1, acc01);
        acc10 = WMMA_F32(a1, b0, acc10);
        acc11 = WMMA_F32(a1, b1, acc11);
    }

    // C/D layout: VGPR r holds row (r + 8*half), col = lane&15
    const bool full = (tile_m * 32 + 32) <= n_nodes;   // wave-uniform
    if (full) {
        #pragma unroll
        for (int r = 0; r < 8; ++r) {
            const int mo = tile_m * 32 + r + 8 * half;
            float* o0 = Out + (size_t)mo * 128 + ncol0 + m;
            float* o1 = Out + (size_t)(mo + 16) * 128 + ncol0 + m;
            o0[0]  = acc00[r];
            o0[16] = acc01[r];
            o1[0]  = acc10[r];
            o1[16] = acc11[r];
        }
    } else {
        #pragma unroll
        for (int r = 0; r < 8; ++r) {
            const int mo = tile_m * 32 + r + 8 * half;
            if (mo < n_nodes) {
                float* o0 = Out + (size_t)mo * 128 + ncol0 + m;
                o0[0]  = acc00[r];
                o0[16] = acc01[r];
            }
            if (mo + 16 < n_nodes) {
                float* o1 = Out + (size_t)(mo + 16) * 128 + ncol0 + m;
                o1[0]  = acc10[r];
                o1[16] = acc11[r];
            }
        }
    }
}

// ---------------------------------------------------------------------------
// 2) per-etype attention scalar:
//    h_e_t[t][head] = sum_d a_e[head][d] * (edge_emb[t] @ W_r_w[t])[head*32+d]
//    (typed never needs materializing; only its a_e-projection is used)
// ---------------------------------------------------------------------------
__global__ void etype_attn_kernel(const float* __restrict__ edge_emb,
                                  const float* __restrict__ W_r_w,
                                  const float* __restrict__ a_e,
                                  float* __restrict__ h_e_t,
                                  int n_etypes) {
    int tid = threadIdx.x;
    if (tid >= n_etypes * HEADS) return;
    int t = tid >> 2, head = tid & 3;
    float s = 0.0f;
    for (int d = 0; d < EDGE_DIM; ++d) {
        float typed = 0.0f;
        for (int e = 0; e < EDGE_DIM; ++e)
            typed += edge_emb[t * EDGE_DIM + e] *
                     W_r_w[((size_t)(t * EDGE_DIM + e)) * (EDGE_DIM * HEADS) +
                           head * EDGE_DIM + d];
        s += a_e[head * EDGE_DIM + d] * typed;
    }
    h_e_t[tid] = s;
}

// ---------------------------------------------------------------------------
// 3) per-(node,head) attention dots: h_l = <a_l, emb>, h_r = <a_r, emb>
// ---------------------------------------------------------------------------
__global__ void node_attn_kernel(const float* __restrict__ emb,
                                 const float* __restrict__ a_l,
                                 const float* __restrict__ a_r,
                                 float* __restrict__ h_l,
                                 float* __restrict__ h_r,
                                 int n_nodes) {
    int idx = blockIdx.x * blockDim.x + threadIdx.x;   // node*4 + head
    if (idx >= n_nodes * HEADS) return;
    int node = idx >> 2, head = idx & 3;
    const float* e = emb + (size_t)node * 128 + head * OUT_DIM;
    float sl = 0.0f, sr = 0.0f;
    #pragma unroll
    for (int d = 0; d < OUT_DIM; ++d) {
        float v = e[d];
        sl += a_l[head * OUT_DIM + d] * v;
        sr += a_r[head * OUT_DIM + d] * v;
    }
    h_l[idx] = sl;
    h_r[idx] = sr;
}

// ---------------------------------------------------------------------------
// 4) edge logits (leaky relu) + segment max over dst (col) via float atomicMax
// ---------------------------------------------------------------------------
__global__ void edge_logits_kernel(const int* __restrict__ row,
                                   const int* __restrict__ col,
                                   const int* __restrict__ etype,
                                   const float* __restrict__ h_l,
                                   const float* __restrict__ h_r,
                                   const float* __restrict__ h_e_t,
                                   float* __restrict__ logits,
                                   float* __restrict__ mbuf,
                                   int n_edges) {
    int idx = blockIdx.x * blockDim.x + threadIdx.x;   // edge*4 + head
    if (idx >= n_edges * HEADS) return;
    int e = idx >> 2, head = idx & 3;
    float lg = h_l[row[e] * HEADS + head] + h_r[col[e] * HEADS + head] +
               h_e_t[etype[e] * HEADS + head];
    lg = (lg > 0.0f) ? lg : lg * NEG_SLOPE;
    logits[idx] = lg;
    atomicMaxF(&mbuf[col[e] * HEADS + head], lg);
}

// ---------------------------------------------------------------------------
// 5) exp(logit - max) + segment sum (den) via float atomicAdd
// ---------------------------------------------------------------------------
__global__ void edge_exp_kernel(const int* __restrict__ col,
                                const float* __restrict__ mbuf,
                                float* __restrict__ exbuf,   // in: logits, out: ex
                                float* __restrict__ den,
                                int n_edges) {
    int idx = blockIdx.x * blockDim.x + threadIdx.x;
    if (idx >= n_edges * HEADS) return;
    int e = idx >> 2, head = idx & 3;
    float ex = expf(exbuf[idx] - mbuf[col[e] * HEADS + head]);
    exbuf[idx] = ex;
    atomicAdd(&den[col[e] * HEADS + head], ex);
}

// ---------------------------------------------------------------------------
// 6) weighted scatter: one wave per edge. Output layout is the transposed
//    one: out[n, d*HEADS + head]. Each lane owns head = lane&3 (alpha hoisted)
//    and the wave's 32 atomics per step cover 128 contiguous bytes.
// ---------------------------------------------------------------------------
__global__ void edge_scatter_kernel(const int* __restrict__ row,
                                    const int* __restrict__ col,
                                    const float* __restrict__ emb,
                                    const float* __restrict__ exbuf,
                                    const float* __restrict__ den,
                                    float* __restrict__ out,
                                    int n_edges) {
    int e = blockIdx.x * (blockDim.x >> 5) + (threadIdx.x >> 5);
    if (e >= n_edges) return;
    int lane = threadIdx.x & 31;
    int r = row[e], c = col[e];
    int head = lane & 3;
    float alpha = exbuf[e * HEADS + head] / den[c * HEADS + head];
    const float* er = emb + (size_t)r * 128 + head * OUT_DIM;
    float*       oc = out + (size_t)c * 128;
    #pragma unroll
    for (int q = 0; q < 4; ++q) {
        int d = q * 8 + (lane >> 2);                  // out col j=q*32+lane -> d=j>>2
        atomicAdd(&oc[q * 32 + lane], er[d] * alpha); // 128B-contiguous atomics
    }
}

// ---------------------------------------------------------------------------
// 7) finalize: out = elu(agg + res + res_b)
// ---------------------------------------------------------------------------
__global__ void finalize_kernel(float* __restrict__ out,
                                const float* __restrict__ res,
                                const float* __restrict__ res_b,
                                int n_nodes) {
    int idx = blockIdx.x * blockDim.x + threadIdx.x;
    if (idx >= n_nodes * 128) return;
    float v = out[idx] + res[idx] + res_b[idx & 127];
    out[idx] = (v > 0.0f) ? v : (expf(v) - 1.0f);
}

// ---------------------------------------------------------------------------
extern "C" void kernel_launch(void* const* d_in, const int* in_sizes, int n_in,
                              void* d_out, int out_size, void* d_ws, size_t ws_size,
                              hipStream_t stream) {
    const float* h        = (const float*)d_in[0];
    const int*   row      = (const int*)  d_in[1];
    const int*   col      = (const int*)  d_in[2];
    const int*   etype    = (const int*)  d_in[3];
    const float* edge_emb = (const float*)d_in[4];
    const float* W        = (const float*)d_in[5];
    const float* W_r_w    = (const float*)d_in[6];
    const float* a_l      = (const float*)d_in[7];
    const float* a_r      = (const float*)d_in[8];
    const float* a_e      = (const float*)d_in[9];
    const float* res_w    = (const float*)d_in[10];
    const float* res_b    = (const float*)d_in[11];
    float*       out      = (float*)d_out;

    const int n_nodes  = in_sizes[0] / IN_DIM;
    const int n_edges  = in_sizes[1];
    const int n_etypes = in_sizes[4] / EDGE_DIM;

    // workspace carve-up (floats)
    float* ws    = (float*)d_ws;
    float* emb   = ws;                                   // N*128
    float* res   = emb   + (size_t)n_nodes * 128;        // N*128
    float* h_l   = res   + (size_t)n_nodes * 128;        // N*4
    float* h_r   = h_l   + (size_t)n_nodes * HEADS;      // N*4
    float* mbuf  = h_r   + (size_t)n_nodes * HEADS;      // N*4
    float* den   = mbuf  + (size_t)n_nodes * HEADS;      // N*4
    float* h_e_t = den   + (size_t)n_nodes * HEADS;      // 64
    float* exbuf = h_e_t + 64;                           // E*4

    const int B = 256;

    // 0) init accumulators
    {
        int n = n_nodes * 128;
        init_kernel<<<(n + B - 1) / B, B, 0, stream>>>(out, mbuf, den, n_nodes);
    }
    // 1) WMMA fp32 projections (2x2 register-tiled)
    {
        dim3 grid((n_nodes + 31) / 32, 2, 1);
        gemm_proj_kernel<<<grid, 128, 0, stream>>>(h, W, res_w, emb, res, n_nodes);
    }
    // 2) per-etype attention scalars
    etype_attn_kernel<<<1, 64, 0, stream>>>(edge_emb, W_r_w, a_e, h_e_t, n_etypes);
    // 3) per-node attention dots
    {
        int n = n_nodes * HEADS;
        node_attn_kernel<<<(n + B - 1) / B, B, 0, stream>>>(emb, a_l, a_r, h_l, h_r, n_nodes);
    }
    // 4) logits + segment max
    {
        int n = n_edges * HEADS;
        edge_logits_kernel<<<(n + B - 1) / B, B, 0, stream>>>(
            row, col, etype, h_l, h_r, h_e_t, exbuf, mbuf, n_edges);
    }
    // 5) exp + segment sum
    {
        int n = n_edges * HEADS;
        edge_exp_kernel<<<(n + B - 1) / B, B, 0, stream>>>(col, mbuf, exbuf, den, n_edges);
    }
    // 6) weighted scatter aggregation (wave per edge, coalesced atomics)
    {
        int wavesPerBlk = B / 32;
        edge_scatter_kernel<<<(n_edges + wavesPerBlk - 1) / wavesPerBlk, B, 0, stream>>>(
            row, col, emb, exbuf, den, out, n_edges);
    }
    // 7) residual + elu
    {
        int n = n_nodes * 128;
        finalize_kernel<<<(n + B - 1) / B, B, 0, stream>>>(out, res, res_b, n_nodes);
    }
}